// Net_24386824307421
// MI455X (gfx1250) — compile-verified
//
#include <hip/hip_runtime.h>
#include <hip/hip_bf16.h>
#include <math.h>

#define BATCH 512
#define N_RES 1510
#define N_DOM 4
#define TAU   0.05f
#define EPS_ANGLE 1e-6f

// workspace float offsets
#define WS_MASK 0          // N_RES*4 = 6040 floats (padded to 6144)
#define WS_AA   6144       // BATCH*N_DOM*3 = 6144 floats
#define WS_TV   12288      // BATCH*N_DOM*3 = 6144 floats
#define WS_DUMP 18432      // spill slot for out-of-range WMMA store lanes (32 floats)

typedef float v2f __attribute__((ext_vector_type(2)));
typedef float v8f __attribute__((ext_vector_type(8)));

#if defined(__has_builtin)
#if __has_builtin(__builtin_amdgcn_wmma_f32_16x16x4_f32)
#define HAVE_WMMA_F32_16X16X4 1
#endif
#endif

// ---------------- Kernel A: mask (R x 4) ----------------
__global__ void mask_kernel(const float* __restrict__ cmeans,
                            const float* __restrict__ cstd,
                            const float* __restrict__ cprop,
                            float* __restrict__ ws_mask,
                            float* __restrict__ out_mask) {
    int r = blockIdx.x * blockDim.x + threadIdx.x;
    if (r >= N_RES) return;
    // log softmax(proportions)
    float pm = cprop[0];
    #pragma unroll
    for (int d = 1; d < N_DOM; ++d) pm = fmaxf(pm, cprop[d]);
    float ps = 0.f;
    #pragma unroll
    for (int d = 0; d < N_DOM; ++d) ps += __expf(cprop[d] - pm);
    float logZ = pm + logf(ps);

    float rf = (float)r;
    float z[N_DOM];
    float zm = -3.4e38f;
    #pragma unroll
    for (int d = 0; d < N_DOM; ++d) {
        float diff = rf - cmeans[d];
        float s = cstd[d];
        float ln = -0.5f * diff * diff / (s * s) + (cprop[d] - logZ);
        z[d] = ln / TAU;
        zm = fmaxf(zm, z[d]);
    }
    float es = 0.f, e[N_DOM];
    #pragma unroll
    for (int d = 0; d < N_DOM; ++d) { e[d] = expf(z[d] - zm); es += e[d]; }
    float inv = 1.f / es;
    #pragma unroll
    for (int d = 0; d < N_DOM; ++d) {
        float m = e[d] * inv;
        ws_mask[r * 4 + d] = m;
        out_mask[r * 4 + d] = m;
    }
}

// ------------- Kernel B: per-(b,d) axis-angle + translation vecs -------------
__global__ void bd_kernel(const float* __restrict__ mean_t,
                          const float* __restrict__ std_t,
                          const float* __restrict__ mean_q,
                          const float* __restrict__ std_q,
                          const float* __restrict__ lf,       // 3x3 row-major
                          const float* __restrict__ eps_t,    // B,D,3
                          const float* __restrict__ eps_q,    // B,D,3
                          float* __restrict__ ws_aa,
                          float* __restrict__ ws_tv) {
    int idx = blockIdx.x * blockDim.x + threadIdx.x;
    if (idx >= BATCH * N_DOM) return;
    int d = idx % N_DOM;

    // quats3 = mean_q + eps_q*std_q ; q = (1, quats3)
    float q1 = mean_q[d * 3 + 0] + eps_q[idx * 3 + 0] * std_q[d * 3 + 0];
    float q2 = mean_q[d * 3 + 1] + eps_q[idx * 3 + 1] * std_q[d * 3 + 1];
    float q3 = mean_q[d * 3 + 2] + eps_q[idx * 3 + 2] * std_q[d * 3 + 2];
    float norms = sqrtf(q1 * q1 + q2 * q2 + q3 * q3);
    float half = atan2f(norms, 1.0f);
    float ang = 2.f * half;
    bool small = fabsf(ang) < EPS_ANGLE;
    float safe = small ? 1.f : ang;
    float sho = small ? (0.5f - ang * ang * (1.f / 48.f)) : (sinf(half) / safe);
    float inv_sho = 1.f / sho;
    ws_aa[idx * 3 + 0] = q1 * inv_sho;
    ws_aa[idx * 3 + 1] = q2 * inv_sho;
    ws_aa[idx * 3 + 2] = q3 * inv_sho;

    // trans_scalars = mean_t + eps_t*std_t ; tv = ts @ lf^T -> tv[k] = sum_j ts[j]*lf[k][j]
    float t0 = mean_t[d * 3 + 0] + eps_t[idx * 3 + 0] * std_t[d * 3 + 0];
    float t1 = mean_t[d * 3 + 1] + eps_t[idx * 3 + 1] * std_t[d * 3 + 1];
    float t2 = mean_t[d * 3 + 2] + eps_t[idx * 3 + 2] * std_t[d * 3 + 2];
    #pragma unroll
    for (int k = 0; k < 3; ++k)
        ws_tv[idx * 3 + k] = t0 * lf[k * 3 + 0] + t1 * lf[k * 3 + 1] + t2 * lf[k * 3 + 2];
}

// ------------- Kernel W: translation_per_residue via V_WMMA_F32_16X16X4_F32 -------------
// One wave = one (batch, 16-residue tile): D(16x16) = mask_tile(16x4) x tv_b(4x16 zero-padded)
// All loads are index-clamped (in-bounds) with value selects -> no EXEC manipulation
// before the WMMA; stores go through cndmask-selected addresses (spill slot for
// out-of-range lanes) -> no EXEC branches at all, EXEC is all-1s throughout.
__global__ void trans_wmma_kernel(const float* __restrict__ ws_mask,
                                  const float* __restrict__ ws_tv,
                                  float* __restrict__ ws_dump,
                                  float* __restrict__ out_trans) {
    const int TILES = (N_RES + 15) / 16;  // 95
    int lane = threadIdx.x & 31;
    int wave = (blockIdx.x * blockDim.x + threadIdx.x) >> 5;
    int b = wave / TILES;
    int tile = wave % TILES;
    if (b >= BATCH) return;  // wave-uniform (grid is exact; defensive)

    // A: 32-bit 16x4 layout — lanes 0-15: (K0,K1), lanes 16-31: (K2,K3), M = lane%16
    int rr = tile * 16 + (lane & 15);
    int rr_c = (rr < N_RES) ? rr : (N_RES - 1);    // clamp: load stays in-bounds
    bool r_ok = (rr < N_RES);
    int klo = (lane < 16) ? 0 : 2;
    float a0 = ws_mask[rr_c * 4 + klo];
    float a1 = ws_mask[rr_c * 4 + klo + 1];
    v2f a;
    a.x = r_ok ? a0 : 0.f;
    a.y = r_ok ? a1 : 0.f;

    // B: 4x16, rows split across lane halves: lanes 0-15 rows {0,1}, lanes 16-31 rows {2,3}
    int nn = lane & 15;
    int nn_c = (nn < 3) ? nn : 2;                  // clamp
    bool n_ok = (nn < 3);
    const float* tvb = ws_tv + b * (N_DOM * 3);
    float b0 = tvb[klo * 3 + nn_c];
    float b1 = tvb[(klo + 1) * 3 + nn_c];
    v2f bm;
    bm.x = n_ok ? b0 : 0.f;
    bm.y = n_ok ? b1 : 0.f;

    int rbase = tile * 16 + ((lane < 16) ? 0 : 8);
    float res[8];
#ifdef HAVE_WMMA_F32_16X16X4
    v8f c = {0.f, 0.f, 0.f, 0.f, 0.f, 0.f, 0.f, 0.f};
    v8f dacc = __builtin_amdgcn_wmma_f32_16x16x4_f32(
        /*neg_a=*/false, a, /*neg_b=*/false, bm,
        /*c_mod=*/(short)0, c, /*reuse_a=*/false, /*reuse_b=*/false);
    #pragma unroll
    for (int v = 0; v < 8; ++v) res[v] = dacc[v];
#else
    #pragma unroll
    for (int v = 0; v < 8; ++v) {
        int r = rbase + v;
        float acc = 0.f;
        if (r < N_RES && nn < 3) {
            #pragma unroll
            for (int d4 = 0; d4 < N_DOM; ++d4)
                acc += ws_mask[r * 4 + d4] * tvb[d4 * 3 + nn];
        }
        res[v] = acc;
    }
#endif
    // D layout: lane holds N = lane%16; VGPR v -> row M = v (lanes 0-15) / v+8 (lanes 16-31)
    // Unconditional stores via selected address: invalid lanes write to ws spill slot.
    float* dump = ws_dump + lane;  // per-lane slot, value is don't-care
    #pragma unroll
    for (int v = 0; v < 8; ++v) {
        int r = rbase + v;
        bool ok = n_ok && (r < N_RES);
        float* addr = ok ? (out_trans + ((size_t)b * N_RES + r) * 3 + nn) : dump;
        *addr = res[v];
    }
}

// ------------- Kernel C: rotations + deform, one thread per (b,r) -------------
__global__ void deform_kernel(const float* __restrict__ lf,
                              const float* __restrict__ pos,      // (3R,3)
                              const float* __restrict__ ws_mask,
                              const float* __restrict__ ws_aa,
                              const float* __restrict__ ws_tv,
                              float* __restrict__ out_atoms) {    // (B,3R,3)
    int idx = blockIdx.x * blockDim.x + threadIdx.x;
    if (idx >= BATCH * N_RES) return;
    int b = idx / N_RES;
    int r = idx - b * N_RES;

    float4 mk = ((const float4*)ws_mask)[r];
    float m4[4] = {mk.x, mk.y, mk.z, mk.w};

    float L[9];
    #pragma unroll
    for (int i = 0; i < 9; ++i) L[i] = lf[i];

    // overall = I
    float O[9] = {1.f, 0.f, 0.f, 0.f, 1.f, 0.f, 0.f, 0.f, 1.f};

    #pragma unroll
    for (int d = 0; d < N_DOM; ++d) {
        const float* aap = ws_aa + ((size_t)b * N_DOM + d) * 3;
        float vx = m4[d] * aap[0];
        float vy = m4[d] * aap[1];
        float vz = m4[d] * aap[2];
        float ang = sqrtf(vx * vx + vy * vy + vz * vz);
        float half = 0.5f * ang;
        bool small = ang < EPS_ANGLE;
        float safe = small ? 1.f : ang;
        float sho = small ? (0.5f - ang * ang * (1.f / 48.f)) : (sinf(half) / safe);
        float qr = cosf(half);
        float qi = vx * sho, qj = vy * sho, qk = vz * sho;
        float two_s = 2.f / (qr * qr + qi * qi + qj * qj + qk * qk);
        float m[9];
        m[0] = 1.f - two_s * (qj * qj + qk * qk);
        m[1] = two_s * (qi * qj - qk * qr);
        m[2] = two_s * (qi * qk + qj * qr);
        m[3] = two_s * (qi * qj + qk * qr);
        m[4] = 1.f - two_s * (qi * qi + qk * qk);
        m[5] = two_s * (qj * qk - qi * qr);
        m[6] = two_s * (qi * qk - qj * qr);
        m[7] = two_s * (qj * qk + qi * qr);
        m[8] = 1.f - two_s * (qi * qi + qj * qj);
        // O = m^T @ O
        float T[9];
        #pragma unroll
        for (int aa2 = 0; aa2 < 3; ++aa2)
            #pragma unroll
            for (int cc = 0; cc < 3; ++cc)
                T[aa2 * 3 + cc] = m[0 * 3 + aa2] * O[0 * 3 + cc]
                                + m[1 * 3 + aa2] * O[1 * 3 + cc]
                                + m[2 * 3 + aa2] * O[2 * 3 + cc];
        #pragma unroll
        for (int i = 0; i < 9; ++i) O[i] = T[i];
    }

    // OL = overall @ lf ; rotated_frame = OL^T, out[k] = sum_j rel[j]*OL[k][j]
    float OL[9];
    #pragma unroll
    for (int k = 0; k < 3; ++k)
        #pragma unroll
        for (int j = 0; j < 3; ++j)
            OL[k * 3 + j] = O[k * 3 + 0] * L[0 * 3 + j]
                          + O[k * 3 + 1] * L[1 * 3 + j]
                          + O[k * 3 + 2] * L[2 * 3 + j];

    // translation for this residue (recomputed inline; out_trans slice written by kernel W)
    const float* tvb = ws_tv + (size_t)b * (N_DOM * 3);
    float tr[3];
    #pragma unroll
    for (int k = 0; k < 3; ++k)
        tr[k] = m4[0] * tvb[0 * 3 + k] + m4[1] * tvb[1 * 3 + k]
              + m4[2] * tvb[2 * 3 + k] + m4[3] * tvb[3 * 3 + k];

    #pragma unroll
    for (int a = 0; a < 3; ++a) {
        int n = 3 * r + a;
        float p0 = pos[n * 3 + 0], p1 = pos[n * 3 + 1], p2 = pos[n * 3 + 2];
        float rel[3];
        #pragma unroll
        for (int j = 0; j < 3; ++j)
            rel[j] = p0 * L[0 * 3 + j] + p1 * L[1 * 3 + j] + p2 * L[2 * 3 + j];
        size_t obase = ((size_t)b * (3 * N_RES) + n) * 3;
        #pragma unroll
        for (int k = 0; k < 3; ++k)
            out_atoms[obase + k] = rel[0] * OL[k * 3 + 0] + rel[1] * OL[k * 3 + 1]
                                 + rel[2] * OL[k * 3 + 2] + tr[k];
    }
}

extern "C" void kernel_launch(void* const* d_in, const int* in_sizes, int n_in,
                              void* d_out, int out_size, void* d_ws, size_t ws_size,
                              hipStream_t stream) {
    const float* cmeans  = (const float*)d_in[1];
    const float* cstd    = (const float*)d_in[2];
    const float* cprop   = (const float*)d_in[3];
    const float* mean_t  = (const float*)d_in[4];
    const float* std_t   = (const float*)d_in[5];
    const float* mean_q  = (const float*)d_in[6];
    const float* std_q   = (const float*)d_in[7];
    const float* lf      = (const float*)d_in[8];
    const float* pos     = (const float*)d_in[9];
    const float* eps_t   = (const float*)d_in[10];
    const float* eps_q   = (const float*)d_in[11];

    float* ws      = (float*)d_ws;
    float* ws_mask = ws + WS_MASK;
    float* ws_aa   = ws + WS_AA;
    float* ws_tv   = ws + WS_TV;
    float* ws_dump = ws + WS_DUMP;

    float* out       = (float*)d_out;
    float* out_atoms = out;                                         // B*3R*3
    float* out_mask  = out + (size_t)BATCH * 3 * N_RES * 3;         // R*4
    float* out_trans = out_mask + (size_t)N_RES * 4;                // B*R*3

    mask_kernel<<<(N_RES + 255) / 256, 256, 0, stream>>>(cmeans, cstd, cprop, ws_mask, out_mask);
    bd_kernel<<<(BATCH * N_DOM + 255) / 256, 256, 0, stream>>>(
        mean_t, std_t, mean_q, std_q, lf, eps_t, eps_q, ws_aa, ws_tv);

    const int TILES = (N_RES + 15) / 16;                 // 95
    int wmma_waves = BATCH * TILES;                      // 48640
    int wmma_blocks = (wmma_waves * 32 + 255) / 256;     // 6080
    trans_wmma_kernel<<<wmma_blocks, 256, 0, stream>>>(ws_mask, ws_tv, ws_dump, out_trans);

    int total = BATCH * N_RES;                           // 773120
    deform_kernel<<<(total + 255) / 256, 256, 0, stream>>>(
        lf, pos, ws_mask, ws_aa, ws_tv, out_atoms);
}